// WindGNN_89292370084484
// MI455X (gfx1250) — compile-verified
//
#include <hip/hip_runtime.h>
#include <hip/hip_bf16.h>

typedef __attribute__((ext_vector_type(2))) float v2f;
typedef __attribute__((ext_vector_type(8))) float v8f;

// ---------------------------------------------------------------------------
// 1) degree / normalization
// ---------------------------------------------------------------------------
__global__ void k_fill_one(float* deg, int n) {
    int i = blockIdx.x * blockDim.x + threadIdx.x;
    if (i < n) deg[i] = 1.0f;  // self-loop contribution
}

__global__ void k_deg_count(const int* __restrict__ dst, float* deg, int e) {
    int i = blockIdx.x * blockDim.x + threadIdx.x;
    if (i < e) atomicAdd(&deg[dst[i]], 1.0f);
}

__global__ void k_rsqrt_inplace(float* deg, int n) {
    int i = blockIdx.x * blockDim.x + threadIdx.x;
    if (i < n) deg[i] = rsqrtf(deg[i]);  // deg >= 1 always (self-loops)
}

// ---------------------------------------------------------------------------
// 2) layer-1 GEMM via V_WMMA_F32_16X16X4_F32  (h1[N,32] = x[N,2] @ W1[2,32])
//    One wave per 16-node tile; K=2 padded to 4 with zeros.
//    A 16x4 f32 layout: lanes 0-15 hold K=0 (v0) / K=1 (v1); lanes 16-31 hold
//    K=2/K=3 (zero padding).  B 4x16 mirrors (rows K striped the same way).
//    C/D: VGPR r, lanes 0-15 -> (M=r, N=lane), lanes 16-31 -> (M=r+8, N=lane-16).
// ---------------------------------------------------------------------------
__global__ void k_gemm1_wmma(const float* __restrict__ x,
                             const float* __restrict__ W1,
                             float* __restrict__ h1,
                             int n_tiles, int n_nodes) {
    int wave = (blockIdx.x * blockDim.x + threadIdx.x) >> 5;
    int lane = threadIdx.x & 31;
    if (wave >= n_tiles) return;  // wave-uniform: EXEC stays all-1 below

    int base = wave * 16;
    int m = lane & 15;
    bool lo = lane < 16;
    bool full_tile = (base + 16 <= n_nodes);  // wave-uniform

    int node = base + m;
    if (node >= n_nodes) node = n_nodes - 1;  // clamp loads (tail tile only)

    v2f a;
    a.x = lo ? x[node * 2 + 0] : 0.0f;  // K=0  (lanes>=16: K=2 pad -> 0)
    a.y = lo ? x[node * 2 + 1] : 0.0f;  // K=1  (lanes>=16: K=3 pad -> 0)

    int rbase = lo ? 0 : 8;

#pragma unroll
    for (int t = 0; t < 2; ++t) {       // two 16-column tiles of the 32 outputs
        int col = t * 16 + m;
        v2f b;
        b.x = lo ? W1[0 * 32 + col] : 0.0f;  // row K=0
        b.y = lo ? W1[1 * 32 + col] : 0.0f;  // row K=1
        v8f c = {};
        c = __builtin_amdgcn_wmma_f32_16x16x4_f32(
                false, a, false, b, (short)0, c, false, false);

        float* colp = h1 + (size_t)(base + rbase) * 32 + t * 16 + m;
        if (full_tile) {
            // hot path: no EXEC juggling, 8 plain global_store_b32
#pragma unroll
            for (int r = 0; r < 8; ++r) colp[r * 32] = c[r];
        } else {
            // tail tile (not taken for N % 16 == 0)
#pragma unroll
            for (int r = 0; r < 8; ++r) {
                if (base + rbase + r < n_nodes) colp[r * 32] = c[r];
            }
        }
    }
}

// ---------------------------------------------------------------------------
// 3) layer-1 aggregation
// ---------------------------------------------------------------------------
__global__ void k_init_agg1(const float* __restrict__ dinv,
                            const float* __restrict__ h1,
                            float* __restrict__ agg1, int total /* = 32*N */) {
    int t = blockIdx.x * blockDim.x + threadIdx.x;
    if (t >= total) return;
    float di = dinv[t >> 5];
    agg1[t] = di * di * h1[t];  // self-loop term
}

__global__ void k_scatter1(const int* __restrict__ src, const int* __restrict__ dst,
                           const float* __restrict__ dinv,
                           const float* __restrict__ h1,
                           float* __restrict__ agg1, int total /* = 32*E */) {
    int t = blockIdx.x * blockDim.x + threadIdx.x;
    if (t >= total) return;
    int e = t >> 5;        // whole wave works on one edge
    int j = t & 31;        // feature -> coalesced loads + coalesced atomics
    int s = src[e];
    int d = dst[e];
    float nrm = dinv[s] * dinv[d];
    atomicAdd(&agg1[d * 32 + j], nrm * h1[s * 32 + j]);
}

// ---------------------------------------------------------------------------
// 4) bias + ReLU + layer-2 GEMV (32->1) + seed output with self-loop & b2
// ---------------------------------------------------------------------------
__global__ void k_layer2_node(const float* __restrict__ agg1,
                              const float* __restrict__ b1,
                              const float* __restrict__ W2,
                              const float* __restrict__ b2,
                              const float* __restrict__ dinv,
                              float* __restrict__ y1,
                              float* __restrict__ out, int n) {
    int i = blockIdx.x * blockDim.x + threadIdx.x;
    if (i >= n) return;
    const float4* row = (const float4*)(agg1 + (size_t)i * 32);
    const float4* bb  = (const float4*)b1;
    const float4* ww  = (const float4*)W2;
    float acc = 0.0f;
#pragma unroll
    for (int q = 0; q < 8; ++q) {
        float4 r = row[q], bv = bb[q], wv = ww[q];
        acc += fmaxf(r.x + bv.x, 0.0f) * wv.x;
        acc += fmaxf(r.y + bv.y, 0.0f) * wv.y;
        acc += fmaxf(r.z + bv.z, 0.0f) * wv.z;
        acc += fmaxf(r.w + bv.w, 0.0f) * wv.w;
    }
    y1[i] = acc;
    float di = dinv[i];
    out[i] = b2[0] + di * di * acc;  // bias + self-loop seed (b2 commutes)
}

// ---------------------------------------------------------------------------
// 5) layer-2 edge scatter
// ---------------------------------------------------------------------------
__global__ void k_scatter2(const int* __restrict__ src, const int* __restrict__ dst,
                           const float* __restrict__ dinv,
                           const float* __restrict__ y1,
                           float* __restrict__ out, int e) {
    int i = blockIdx.x * blockDim.x + threadIdx.x;
    if (i >= e) return;
    int s = src[i];
    int d = dst[i];
    atomicAdd(&out[d], dinv[s] * dinv[d] * y1[s]);
}

// ---------------------------------------------------------------------------
extern "C" void kernel_launch(void* const* d_in, const int* in_sizes, int n_in,
                              void* d_out, int out_size, void* d_ws, size_t ws_size,
                              hipStream_t stream) {
    const float* x  = (const float*)d_in[0];      // [N,2]
    const int*   ei = (const int*)d_in[1];        // [2,E] -> src=ei, dst=ei+E
    const float* W1 = (const float*)d_in[2];      // [2,32]
    const float* b1 = (const float*)d_in[3];      // [32]
    const float* W2 = (const float*)d_in[4];      // [32,1]
    const float* b2 = (const float*)d_in[5];      // [1]
    float* out = (float*)d_out;                   // [N,1]

    const int N = in_sizes[0] / 2;
    const int E = in_sizes[1] / 2;
    const int* src = ei;
    const int* dst = ei + E;

    // workspace layout (floats): dinv[N] | h1[32N] | agg1[32N] | y1[N]
    float* dinv = (float*)d_ws;
    float* h1   = dinv + N;
    float* agg1 = h1 + (size_t)N * 32;
    float* y1   = agg1 + (size_t)N * 32;

    const int B = 256;
    auto blocks = [](long long work, int b) { return (int)((work + b - 1) / b); };

    // degrees -> dinv
    k_fill_one<<<blocks(N, B), B, 0, stream>>>(dinv, N);
    k_deg_count<<<blocks(E, B), B, 0, stream>>>(dst, dinv, E);
    k_rsqrt_inplace<<<blocks(N, B), B, 0, stream>>>(dinv, N);

    // layer 1
    int n_tiles = (N + 15) / 16;
    k_gemm1_wmma<<<blocks((long long)n_tiles * 32, B), B, 0, stream>>>(x, W1, h1, n_tiles, N);
    k_init_agg1<<<blocks((long long)N * 32, B), B, 0, stream>>>(dinv, h1, agg1, N * 32);
    k_scatter1<<<blocks((long long)E * 32, B), B, 0, stream>>>(src, dst, dinv, h1, agg1, E * 32);

    // layer 2
    k_layer2_node<<<blocks(N, B), B, 0, stream>>>(agg1, b1, W2, b2, dinv, y1, out, N);
    k_scatter2<<<blocks(E, B), B, 0, stream>>>(src, dst, dinv, y1, out, E);
}